// ATSSSSD512Loss_17575006175949
// MI455X (gfx1250) — compile-verified
//
#include <hip/hip_runtime.h>
#include <math.h>

#define BATCH 64
#define NOBJ  32
#define NPRI  5456
#define NCLS  80
#define NLVL  5
#define NCAND 9
#define NENT  45            // NLVL * NCAND
#define EPSF  1e-7f
#define FALPHA 0.25f
#define NROWS (BATCH * NPRI)            // 349184, divisible by 256
#define FBLK  (NROWS / 256)             // 1364
#define CBLK  12                        // covers BATCH*NENT = 2880

typedef __attribute__((ext_vector_type(2))) float v2f;
typedef __attribute__((ext_vector_type(8))) float v8f;

__constant__ int cOFFS[6] = {0, 4096, 5120, 5376, 5440, 5456};
__constant__ int cSIZE[5] = {4096, 1024, 256, 64, 16};

// ---------------------------------------------------------------- zero labels
__global__ void zero_i32(int* p, int n) {
    int i = blockIdx.x * 256 + threadIdx.x;
    if (i < n) p[i] = 0;
}

// --------------------------------------------- WMMA distance + top-9 + cand IoU
// grid = (NLVL, BATCH), block = 256 (8 waves).
// Each wave owns one 16-obj x 16-prior tile of the 32 x 64 chunk:
//   objHalf = wave & 1, tileCol = wave >> 1.
// d^2(g,p) = dot([gx, gy, |g|^2, 1], [-2px, -2py, 1, |p|^2])  -> exact f32 K=4 WMMA.
__global__ void atss_topk_kernel(const float* __restrict__ boxes,
                                 const float* __restrict__ priors,
                                 int* __restrict__ idxArr,
                                 float* __restrict__ povArr) {
    __shared__ float sdist[NOBJ * 65];   // 65-stride: conflict-free column scans

    const int lvl  = blockIdx.x;
    const int b    = blockIdx.y;
    const int loff = cOFFS[lvl];
    const int lsz  = cSIZE[lvl];

    const int tid     = threadIdx.x;
    const int wave    = tid >> 5;
    const int lane    = tid & 31;
    const int objHalf = wave & 1;
    const int tileCol = wave >> 1;      // 0..3
    const int nlane   = lane & 15;

    // ---- A matrix (16x4 f32): per-object features, constant for the block
    // V0: lanes0-15 K=0 (gx), lanes16-31 K=2 (|g|^2); V1: K=1 (gy) / K=3 (1)
    const int om = nlane + objHalf * 16;
    const float* bxp = boxes + ((size_t)b * NOBJ + om) * 4;
    const float gx = 0.5f * (bxp[0] + bxp[2]);
    const float gy = 0.5f * (bxp[1] + bxp[3]);
    v2f Av;
    if (lane < 16) { Av.x = gx;                Av.y = gy;   }
    else           { Av.x = gx * gx + gy * gy; Av.y = 1.0f; }

    // top-9 state (register-resident, threads 0..31 use it), sorted ascending
    float tv[NCAND];
    int   ti[NCAND];
#pragma unroll
    for (int k = 0; k < NCAND; ++k) { tv[k] = 3.0e38f; ti[k] = 0; }

    const int nChunks = (lsz + 63) >> 6;
    for (int ch = 0; ch < nChunks; ++ch) {
        // ---- B matrix (4x16 f32): prior features for this wave's 16 columns
        // lanes0-15: V0 = K0 (-2px), V1 = K1 (-2py); lanes16-31: V0 = K2 (1), V1 = K3 (|p|^2)
        int pj = ch * 64 + tileCol * 16 + nlane;
        if (pj >= lsz) pj = lsz - 1;            // clamp (small last level); ignored by scan
        const float* pp = priors + (size_t)(loff + pj) * 4;
        const float px = pp[0], py = pp[1];
        v2f Bv;
        if (lane < 16) { Bv.x = -2.0f * px; Bv.y = -2.0f * py;        }
        else           { Bv.x = 1.0f;       Bv.y = px * px + py * py; }

        v8f Cz = {0.f, 0.f, 0.f, 0.f, 0.f, 0.f, 0.f, 0.f};
        v8f Dv = __builtin_amdgcn_wmma_f32_16x16x4_f32(
            /*neg_a=*/false, Av, /*neg_b=*/false, Bv,
            /*c_mod=*/(short)0, Cz, /*reuse_a=*/false, /*reuse_b=*/false);

        // D layout: VGPR r -> M=r (lanes0-15) / M=r+8 (lanes16-31), N = lane&15
        const int mbase = ((lane < 16) ? 0 : 8) + objHalf * 16;
        const int col   = tileCol * 16 + nlane;
#pragma unroll
        for (int r = 0; r < 8; ++r)
            sdist[(mbase + r) * 65 + col] = Dv[r];

        __syncthreads();

        if (tid < NOBJ) {                       // one selection thread per object
            int cc = lsz - ch * 64; if (cc > 64) cc = 64;
            const float* drow = &sdist[tid * 65];
            for (int j = 0; j < cc; ++j) {
                float vv = drow[j];
                // early-out: tv[] sorted ascending, tv[8] is the current 9th-best;
                // vv >= tv[8] can never insert (strict '<' also encodes the
                // top_k tie rule: equal distance -> earlier index wins)
                if (vv < tv[NCAND - 1]) {
                    int ii = ch * 64 + j;
#pragma unroll
                    for (int k = 0; k < NCAND; ++k) {
                        if (vv < tv[k]) {
                            float t0 = tv[k]; tv[k] = vv; vv = t0;
                            int   i0 = ti[k]; ti[k] = ii; ii = i0;
                        }
                    }
                }
            }
        }
        __syncthreads();
    }

    // ---- IoU of the 9 candidates vs this object's GT box
    if (tid < NOBJ) {
        const int o = tid;
        const float* bb = boxes + ((size_t)b * NOBJ + o) * 4;
        const float ax0 = bb[0], ay0 = bb[1], ax1 = bb[2], ay1 = bb[3];
        const float areaA = (ax1 - ax0) * (ay1 - ay0);
#pragma unroll
        for (int c = 0; c < NCAND; ++c) {
            const int pi = loff + ti[c];
            const float* pr = priors + (size_t)pi * 4;
            const float hw = pr[2] * 0.5f, hh = pr[3] * 0.5f;
            const float bx0 = pr[0] - hw, by0 = pr[1] - hh;
            const float bx1 = pr[0] + hw, by1 = pr[1] + hh;
            float iw = fminf(ax1, bx1) - fmaxf(ax0, bx0); iw = fmaxf(iw, 0.f);
            float ih = fminf(ay1, by1) - fmaxf(ay0, by0); ih = fmaxf(ih, 0.f);
            const float inter = iw * ih;
            const float areaB = (bx1 - bx0) * (by1 - by0);
            const float iou = inter / (areaA + areaB - inter + EPSF);
            const size_t base = ((size_t)(b * NOBJ + o) * NLVL + lvl) * NCAND + c;
            idxArr[base] = ti[c];
            povArr[base] = iou;
        }
    }
}

// ----------------------------------------------- sequential per-image assignment
__global__ void atss_assign_kernel(const float* __restrict__ boxes,
                                   const int* __restrict__ labels,
                                   const float* __restrict__ locs,
                                   const float* __restrict__ priors,
                                   const int* __restrict__ idxArr,
                                   const float* __restrict__ povArr,
                                   int* __restrict__ lab,
                                   float* __restrict__ tb,
                                   float* __restrict__ db,
                                   int* __restrict__ mk) {
    const int b = blockIdx.x * blockDim.x + threadIdx.x;
    if (b >= BATCH) return;

    // threshold = mean + std(ddof=1) over the 45 candidate IoUs
    float thr[NOBJ];
    for (int o = 0; o < NOBJ; ++o) {
        const float* pv = povArr + (size_t)(b * NOBJ + o) * NENT;
        float s = 0.f;
        for (int k = 0; k < NENT; ++k) s += pv[k];
        const float m = s / 45.0f;
        float v = 0.f;
        for (int k = 0; k < NENT; ++k) { float d = pv[k] - m; v += d * d; }
        thr[o] = m + sqrtf(v / 44.0f);
    }

    for (int l = 0; l < NLVL; ++l) {
        const int loff = cOFFS[l];
        for (int c = 0; c < NCAND; ++c) {
            float bestv = -1.0f; int ob = 0;
            for (int o = 0; o < NOBJ; ++o) {
                const size_t base = ((size_t)(b * NOBJ + o) * NLVL + l) * NCAND + c;
                const float pv = povArr[base];
                const int   id = idxArr[base];
                const float* pp = priors + (size_t)(loff + id) * 4;
                const float* bx = boxes + ((size_t)b * NOBJ + o) * 4;
                const bool inside = (bx[0] < pp[0]) && (pp[0] < bx[2]) &&
                                    (bx[1] < pp[1]) && (pp[1] < bx[3]);
                const float iouc = ((pv > thr[o]) && inside) ? pv : 0.0f;
                if (iouc > bestv) { bestv = iouc; ob = o; }   // first max wins
            }
            const bool match = bestv > 0.0f;
            const size_t base = ((size_t)(b * NOBJ + ob) * NLVL + l) * NCAND + c;
            const int pidx = loff + idxArr[base];
            if (match) lab[(size_t)b * NPRI + pidx] = labels[b * NOBJ + ob];

            const int e = l * NCAND + c;
            const float* bx = boxes + ((size_t)b * NOBJ + ob) * 4;
            float* T = tb + ((size_t)b * NENT + e) * 4;
            T[0] = bx[0]; T[1] = bx[1]; T[2] = bx[2]; T[3] = bx[3];

            const float* L  = locs + ((size_t)b * NPRI + pidx) * 4;
            const float* pr = priors + (size_t)pidx * 4;
            const float cx = L[0] * pr[2] * 0.1f + pr[0];
            const float cy = L[1] * pr[3] * 0.1f + pr[1];
            const float wd = expf(L[2] * 0.2f) * pr[2];
            const float ht = expf(L[3] * 0.2f) * pr[3];
            float* D = db + ((size_t)b * NENT + e) * 4;
            D[0] = cx - wd * 0.5f; D[1] = cy - ht * 0.5f;
            D[2] = cx + wd * 0.5f; D[3] = cy + ht * 0.5f;

            mk[b * NENT + e] = match ? 1 : 0;
        }
    }
}

// --------------------------------------------------- focal loss (memory-bound)
__global__ void focal_kernel(const float* __restrict__ scores,
                             const int* __restrict__ lab,
                             float* __restrict__ fPart,
                             float* __restrict__ nPart) {
    __shared__ float s1[256], s2[256];
    const int t   = threadIdx.x;
    const int row = blockIdx.x * 256 + t;       // grid exactly covers NROWS

    const int tg = lab[row];
    float np = (tg > 0) ? 1.0f : 0.0f;
    float loss = 0.f;

    const float4* S = (const float4*)(scores + (size_t)row * NCLS);
#pragma unroll 4
    for (int q = 0; q < NCLS / 4; ++q) {
        const float4 f = S[q];
        const float xs[4] = {f.x, f.y, f.z, f.w};
#pragma unroll
        for (int k = 0; k < 4; ++k) {
            const float x = xs[k];
            const int cls = q * 4 + k + 1;                 // 1-based class
            const float e   = expf(-fabsf(x));
            const float l1p = log1pf(e);
            const float spx = fmaxf(x, 0.f) + l1p;         // softplus(x)  = -log_sigmoid(-x)
            const float spn = fmaxf(-x, 0.f) + l1p;        // softplus(-x) = -log_sigmoid(x)
            const float p  = 1.0f / (1.0f + expf(-x));
            const float q1 = 1.0f - p;
            loss += (tg == cls) ? (FALPHA * q1 * q1 * spn)
                                : ((1.0f - FALPHA) * p * p * spx);
        }
    }

    s1[t] = loss; s2[t] = np;
    __syncthreads();
    for (int s = 128; s > 0; s >>= 1) {
        if (t < s) { s1[t] += s1[t + s]; s2[t] += s2[t + s]; }
        __syncthreads();
    }
    if (t == 0) { fPart[blockIdx.x] = s1[0]; nPart[blockIdx.x] = s2[0]; }
}

// ------------------------------------------------------------------ CIoU loss
__global__ void ciou_kernel(const float* __restrict__ tb,
                            const float* __restrict__ db,
                            const int* __restrict__ mk,
                            float* __restrict__ lPart,
                            float* __restrict__ mPart) {
    __shared__ float s1[256], s2[256];
    const int t = threadIdx.x;
    const int e = blockIdx.x * 256 + t;
    float ls = 0.f, ms = 0.f;
    if (e < BATCH * NENT) {
        const float* p = db + (size_t)e * 4;
        const float* g = tb + (size_t)e * 4;
        const float mf = (float)mk[e];
        const float pw = p[2] - p[0], ph = p[3] - p[1];
        const float tw = g[2] - g[0], th = g[3] - g[1];
        float iw = fminf(p[2], g[2]) - fmaxf(p[0], g[0]); iw = fmaxf(iw, 0.f);
        float ih = fminf(p[3], g[3]) - fmaxf(p[1], g[1]); ih = fmaxf(ih, 0.f);
        const float inter = iw * ih;
        const float uni   = pw * ph + tw * th - inter;
        const float iou   = inter / (uni + EPSF);
        const float cw  = fmaxf(p[2], g[2]) - fminf(p[0], g[0]);
        const float chh = fmaxf(p[3], g[3]) - fminf(p[1], g[1]);
        const float c2 = cw * cw + chh * chh + EPSF;
        const float dx = p[0] + p[2] - g[0] - g[2];
        const float dy = p[1] + p[3] - g[1] - g[3];
        const float rho2 = (dx * dx + dy * dy) * 0.25f;
        const float dif = atanf(tw / (th + EPSF)) - atanf(pw / (ph + EPSF));
        const float v = (4.0f / (float)(M_PI * M_PI)) * dif * dif;
        const float a = v / (1.0f - iou + v + EPSF);
        const float loss = 1.0f - iou + rho2 / c2 + a * v;
        ls = loss * mf; ms = mf;
    }
    s1[t] = ls; s2[t] = ms;
    __syncthreads();
    for (int s = 128; s > 0; s >>= 1) {
        if (t < s) { s1[t] += s1[t + s]; s2[t] += s2[t + s]; }
        __syncthreads();
    }
    if (t == 0) { lPart[blockIdx.x] = s1[0]; mPart[blockIdx.x] = s2[0]; }
}

// ------------------------------------------------------------- final combine
__global__ void final_kernel(const float* __restrict__ fPart,
                             const float* __restrict__ nPart,
                             const float* __restrict__ lPart,
                             const float* __restrict__ mPart,
                             float* __restrict__ out) {
    if (threadIdx.x == 0 && blockIdx.x == 0) {
        float fs = 0.f, ns = 0.f;
        for (int i = 0; i < FBLK; ++i) { fs += fPart[i]; ns += nPart[i]; }
        float ls = 0.f, ms = 0.f;
        for (int i = 0; i < CBLK; ++i) { ls += lPart[i]; ms += mPart[i]; }
        const float conf = fs / fmaxf(ns, 1.0f);
        const float loc  = ls / fmaxf(ms, 1.0f);
        out[0] = conf + loc;
    }
}

// ------------------------------------------------------------------- launch
extern "C" void kernel_launch(void* const* d_in, const int* in_sizes, int n_in,
                              void* d_out, int out_size, void* d_ws, size_t ws_size,
                              hipStream_t stream) {
    const float* locs   = (const float*)d_in[0];   // (B, P, 4)
    const float* scores = (const float*)d_in[1];   // (B, P, 80)
    const float* boxes  = (const float*)d_in[2];   // (B, 32, 4)
    const int*   labels = (const int*)d_in[3];     // (B, 32)
    const float* priors = (const float*)d_in[4];   // (P, 4)
    float* out = (float*)d_out;

    char* w = (char*)d_ws;
    size_t off = 0;
    int*   lab    = (int*)(w + off);   off += sizeof(int)   * (size_t)NROWS;
    int*   idxArr = (int*)(w + off);   off += sizeof(int)   * (size_t)BATCH * NOBJ * NENT;
    float* povArr = (float*)(w + off); off += sizeof(float) * (size_t)BATCH * NOBJ * NENT;
    float* tb     = (float*)(w + off); off += sizeof(float) * (size_t)BATCH * NENT * 4;
    float* db     = (float*)(w + off); off += sizeof(float) * (size_t)BATCH * NENT * 4;
    int*   mk     = (int*)(w + off);   off += sizeof(int)   * (size_t)BATCH * NENT;
    float* fPart  = (float*)(w + off); off += sizeof(float) * FBLK;
    float* nPart  = (float*)(w + off); off += sizeof(float) * FBLK;
    float* lPart  = (float*)(w + off); off += sizeof(float) * CBLK;
    float* mPart  = (float*)(w + off); off += sizeof(float) * CBLK;
    (void)ws_size; (void)in_sizes; (void)n_in; (void)out_size;

    zero_i32<<<(NROWS + 255) / 256, 256, 0, stream>>>(lab, NROWS);
    atss_topk_kernel<<<dim3(NLVL, BATCH), 256, 0, stream>>>(boxes, priors, idxArr, povArr);
    atss_assign_kernel<<<1, 64, 0, stream>>>(boxes, labels, locs, priors,
                                             idxArr, povArr, lab, tb, db, mk);
    focal_kernel<<<FBLK, 256, 0, stream>>>(scores, lab, fPart, nPart);
    ciou_kernel<<<CBLK, 256, 0, stream>>>(tb, db, mk, lPart, mPart);
    final_kernel<<<1, 1, 0, stream>>>(fPart, nPart, lPart, mPart, out);
}